// GCN_8916352107095
// MI455X (gfx1250) — compile-verified
//
#include <hip/hip_runtime.h>

typedef float v2f __attribute__((ext_vector_type(2)));
typedef float v8f __attribute__((ext_vector_type(8)));

#define H_DIM 256
#define G_DIM 256
#define O_DIM 32
#define R1_DIM 128  // H/2

// ---- order-preserving float <-> uint for atomicMax on floats ----
__device__ __forceinline__ unsigned ordOf(float f) {
  unsigned u = __float_as_uint(f);
  return (u & 0x80000000u) ? ~u : (u | 0x80000000u);
}
__device__ __forceinline__ float ordInv(unsigned o) {
  return (o & 0x80000000u) ? __uint_as_float(o ^ 0x80000000u)
                           : __uint_as_float(~o);
}

// ---- gfx1250 async global->LDS copy (16B per lane), ASYNCcnt tracked ----
__device__ __forceinline__ void async_copy_b128(const float* gp, void* lds_ptr) {
  unsigned lds_off = (unsigned)(unsigned long long)lds_ptr;  // low 32b = AS(3) offset
  asm volatile("global_load_async_to_lds_b128 %0, %1, off"
               :: "v"(lds_off), "v"((unsigned long long)gp)
               : "memory");
}
__device__ __forceinline__ void wait_async_zero() {
  asm volatile("s_wait_asynccnt 0x0" ::: "memory");
}

// ---- degree / symmetric norm (self-loop included via init=1) ----
__global__ void k_deg_init(unsigned* deg, int n) {
  int i = blockIdx.x * blockDim.x + threadIdx.x;
  if (i < n) deg[i] = 1u;
}
__global__ void k_deg_accum(const int* __restrict__ dst, unsigned* deg, int e) {
  int i = blockIdx.x * blockDim.x + threadIdx.x;
  if (i < e) atomicAdd(&deg[dst[i]], 1u);
}
__global__ void k_dis(const unsigned* __restrict__ degu, float* dis, int n) {
  int i = blockIdx.x * blockDim.x + threadIdx.x;
  if (i < n) dis[i] = rsqrtf((float)degu[i]);  // deg >= 1 always (self loop)
}

// ---- fp32 WMMA GEMM: C[M x 256] = A[M x K] @ B[K x 256] (+ bias) ----
// block = 256 threads = 8 waves; block tile = 128 rows x 64 cols.
// Double-buffered LDS tiles filled by async global->LDS copies that overlap
// the 64 v_wmma_f32_16x16x4_f32 of the current K-chunk.
template <int K>
__global__ __launch_bounds__(256) void k_gemm(const float* __restrict__ Ag,
                                              const float* __restrict__ Bg,
                                              const float* __restrict__ bias,
                                              float* __restrict__ Cg, int M) {
  const int Nn = H_DIM;
  __shared__ __align__(16) float As[2][128][68];  // stride 68: 16B-aligned rows,
  __shared__ __align__(16) float Bs[2][64][68];   // conflict-free A-frag reads
  const int tid = threadIdx.x;
  const int lane = tid & 31, wv = tid >> 5;
  const int half = lane >> 4, l16 = lane & 15;
  const int m0 = blockIdx.x * 128;
  const int n0 = blockIdx.y * 64;

  v8f zero = {0.f, 0.f, 0.f, 0.f, 0.f, 0.f, 0.f, 0.f};
  v8f acc[4];
#pragma unroll
  for (int t = 0; t < 4; ++t) acc[t] = zero;

  auto load_chunk = [&](int k0, int buf) {
    // A tile: 128x64 floats = 2048 b128 copies, coalesced 256B runs per row
#pragma unroll
    for (int i = 0; i < 8; ++i) {
      int idx = i * 256 + tid;
      int r = idx >> 4, c4 = idx & 15;
      int gr = m0 + r;
      if (gr > M - 1) gr = M - 1;  // clamp OOB rows (stores are guarded)
      async_copy_b128(Ag + (size_t)gr * K + k0 + c4 * 4, &As[buf][r][c4 * 4]);
    }
    // B tile: 64x64 floats
#pragma unroll
    for (int i = 0; i < 4; ++i) {
      int idx = i * 256 + tid;
      int r = idx >> 4, c4 = idx & 15;
      async_copy_b128(Bg + (size_t)(k0 + r) * Nn + n0 + c4 * 4,
                      &Bs[buf][r][c4 * 4]);
    }
  };

  const int nch = K / 64;
  load_chunk(0, 0);
  for (int ch = 0; ch < nch; ++ch) {
    wait_async_zero();   // this wave's copies for chunk `ch` are in LDS
    __syncthreads();     // ... and everyone else's too; prior compute done
    if (ch + 1 < nch) load_chunk((ch + 1) * 64, (ch + 1) & 1);  // overlap
    const int buf = ch & 1;
    const float(*Ab)[68] = As[buf];
    const float(*Bb)[68] = Bs[buf];
#pragma unroll
    for (int kk = 0; kk < 16; ++kk) {
      const int ka = kk * 4 + half * 2;  // A frag: lanes<16 -> K 0,1; >=16 -> K 2,3
      v2f a;
      a.x = Ab[wv * 16 + l16][ka];
      a.y = Ab[wv * 16 + l16][ka + 1];
#pragma unroll
      for (int t = 0; t < 4; ++t) {
        v2f b;  // B frag: same K split, N striped across lanes
        b.x = Bb[ka][t * 16 + l16];
        b.y = Bb[ka + 1][t * 16 + l16];
        acc[t] = __builtin_amdgcn_wmma_f32_16x16x4_f32(
            false, a, false, b, (short)0, acc[t], false, false);
      }
    }
  }
  // C/D layout: VGPR v -> row v (lanes 0-15) / row v+8 (lanes 16-31)
  const int rbase = m0 + wv * 16 + half * 8;
#pragma unroll
  for (int t = 0; t < 4; ++t) {
    const int col = n0 + t * 16 + l16;
    const float bv = bias ? bias[col] : 0.f;
#pragma unroll
    for (int v = 0; v < 8; ++v) {
      int r = rbase + v;
      if (r < M) Cg[(size_t)r * Nn + col] = acc[t][v] + bv;
    }
  }
}

// ---- h_out[i] = conv_b + hw[i] * dis[i]^2  (self-loop term + bias folded) ----
__global__ void k_selfloop_bias(const float* __restrict__ hw,
                                const float* __restrict__ dis,
                                const float* __restrict__ bias,
                                float* __restrict__ hout, int n) {
  int node = blockIdx.x, f = threadIdx.x;
  float di = dis[node];
  hout[(size_t)node * H_DIM + f] = bias[f] + hw[(size_t)node * H_DIM + f] * di * di;
}

// ---- edge scatter-add: one block per edge, one thread per feature ----
// hw and hout are L2-resident (51 MB << 192 MB L2): gather hits L2, the
// float atomic-add RMW happens at the L2 atomic units.
__global__ void k_edge_scatter(const int* __restrict__ src,
                               const int* __restrict__ dst,
                               const float* __restrict__ dis,
                               const float* __restrict__ hw,
                               float* __restrict__ hout, int e) {
  int ed = blockIdx.x;
  if (ed >= e) return;
  int s = src[ed], d = dst[ed];
  float w = dis[s] * dis[d];
  int f = threadIdx.x;
  atomicAdd(&hout[(size_t)d * H_DIM + f], hw[(size_t)s * H_DIM + f] * w);
}

// ---- pooling ----
__global__ void k_pool_zero(float* sums, unsigned* maxo, unsigned* cnts) {
  int i = blockIdx.x * blockDim.x + threadIdx.x;
  if (i < G_DIM * H_DIM) {
    sums[i] = 0.f;
    maxo[i] = 0u;  // below ordOf(-inf)=0x007FFFFF; only read when count>0
  }
  if (i < G_DIM) cnts[i] = 0u;
}
__global__ void k_pool(const float* __restrict__ h, const int* __restrict__ batch,
                       float* sums, unsigned* maxo, unsigned* cnts, int n) {
  int node = blockIdx.x, f = threadIdx.x;
  int g = batch[node];
  float v = h[(size_t)node * H_DIM + f];
  atomicAdd(&sums[g * H_DIM + f], v);
  atomicMax(&maxo[g * H_DIM + f], ordOf(v));
  if (f == 0) atomicAdd(&cnts[g], 1u);
}

// ---- readout MLP: [sum|max|mean] -> leaky_relu(Linear) -> Linear ----
__global__ __launch_bounds__(128) void k_readout(
    const float* __restrict__ sums, const unsigned* __restrict__ maxo,
    const unsigned* __restrict__ cnts, const float* __restrict__ r1w,
    const float* __restrict__ r1b, const float* __restrict__ r2w,
    const float* __restrict__ r2b, float* __restrict__ out) {
  int g = blockIdx.x, tid = threadIdx.x;
  __shared__ float hg[3 * H_DIM];
  __shared__ float z[R1_DIM];
  unsigned c = cnts[g];
  float inv = 1.0f / (float)(c > 0u ? c : 1u);
  for (int j = tid; j < H_DIM; j += 128) {
    float s = sums[g * H_DIM + j];
    hg[j] = s;
    hg[H_DIM + j] = (c > 0u) ? ordInv(maxo[g * H_DIM + j]) : 0.f;
    hg[2 * H_DIM + j] = s * inv;
  }
  __syncthreads();
  float a = r1b[tid];
  for (int k = 0; k < 3 * H_DIM; ++k) a += hg[k] * r1w[k * R1_DIM + tid];
  z[tid] = (a > 0.f) ? a : 0.01f * a;
  __syncthreads();
  if (tid < O_DIM) {
    float o = r2b[tid];
    for (int j = 0; j < R1_DIM; ++j) o += z[j] * r2w[j * O_DIM + tid];
    out[g * O_DIM + tid] = o;
  }
}

extern "C" void kernel_launch(void* const* d_in, const int* in_sizes, int n_in,
                              void* d_out, int out_size, void* d_ws,
                              size_t ws_size, hipStream_t stream) {
  const float* x      = (const float*)d_in[0];
  const int*   eidx   = (const int*)d_in[1];
  const int*   batch  = (const int*)d_in[2];
  const float* enc_w  = (const float*)d_in[3];
  const float* enc_b  = (const float*)d_in[4];
  const float* conv_w = (const float*)d_in[5];
  const float* conv_b = (const float*)d_in[6];
  const float* r1w    = (const float*)d_in[7];
  const float* r1b    = (const float*)d_in[8];
  const float* r2w    = (const float*)d_in[9];
  const float* r2b    = (const float*)d_in[10];
  float* out = (float*)d_out;

  const int N = in_sizes[2];      // 50000 nodes (batch has N elems)
  const int E = in_sizes[1] / 2;  // edge_index is [2, E]
  const int* src = eidx;
  const int* dst = eidx + E;

  char* ws = (char*)d_ws;
  auto align_up = [](size_t v) { return (v + 255) & ~(size_t)255; };
  size_t off = 0;
  float*    dis  = (float*)(ws + off);    off += align_up((size_t)N * 4);
  float*    hA   = (float*)(ws + off);    off += align_up((size_t)N * H_DIM * 4);
  float*    hB   = (float*)(ws + off);    off += align_up((size_t)N * H_DIM * 4);
  float*    sums = (float*)(ws + off);    off += align_up((size_t)G_DIM * H_DIM * 4);
  unsigned* maxo = (unsigned*)(ws + off); off += align_up((size_t)G_DIM * H_DIM * 4);
  unsigned* cnts = (unsigned*)(ws + off); off += align_up((size_t)G_DIM * 4);

  // 1) degree + symmetric norm (deg buffer aliases dis, rewritten in place)
  k_deg_init<<<(N + 255) / 256, 256, 0, stream>>>((unsigned*)dis, N);
  k_deg_accum<<<(E + 255) / 256, 256, 0, stream>>>(dst, (unsigned*)dis, E);
  k_dis<<<(N + 255) / 256, 256, 0, stream>>>((const unsigned*)dis, dis, N);

  dim3 gg((N + 127) / 128, H_DIM / 64);
  // 2) encoder: hA = x @ enc_w + enc_b   (K = 128)
  k_gemm<128><<<gg, 256, 0, stream>>>(x, enc_w, enc_b, hA, N);

  // 3) three GCN layers
  for (int l = 0; l < 3; ++l) {
    k_gemm<256><<<gg, 256, 0, stream>>>(
        hA, conv_w + (size_t)l * H_DIM * H_DIM, nullptr, hB, N);
    k_selfloop_bias<<<N, H_DIM, 0, stream>>>(hB, dis, conv_b + l * H_DIM, hA, N);
    k_edge_scatter<<<E, H_DIM, 0, stream>>>(src, dst, dis, hB, hA, E);
  }

  // 4) graph pooling + readout MLP
  k_pool_zero<<<(G_DIM * H_DIM + 255) / 256, 256, 0, stream>>>(sums, maxo, cnts);
  k_pool<<<N, H_DIM, 0, stream>>>(hA, batch, sums, maxo, cnts, N);
  k_readout<<<G_DIM, 128, 0, stream>>>(sums, maxo, cnts, r1w, r1b, r2w, r2b, out);

  (void)ws_size; (void)n_in; (void)out_size;
}